// CSPN_original_16200616640826
// MI455X (gfx1250) — compile-verified
//
#include <hip/hip_runtime.h>
#include <hip/hip_bf16.h>
#include <stdint.h>

// CSPN 3x3 propagation step, memory-bound streaming kernel for MI455X.
// Traffic-optimal plan: ~164 MB moved once -> ~7us at 23.3 TB/s.
// CDNA5 paths used: global_load_async_to_lds_b32 (ASYNCcnt) for the input
// halo tile, non-temporal b128 loads for the 9 single-use weight planes,
// non-temporal b128 store for the output.

#define TW 128              // tile width  (outputs per block in x)
#define TH 8                // tile height (outputs per block in y)
#define HALO_W (TW + 2)     // 130
#define HALO_H (TH + 2)     // 10
#define LP 132              // padded LDS row stride (floats)

typedef float v4f __attribute__((ext_vector_type(4)));

__device__ __forceinline__ void async_ld_b32_to_lds(unsigned lds_byte_off,
                                                    const float* gaddr) {
  // GLOBAL_LOAD_ASYNC_TO_LDS_B32 (GV mode): per-lane LDS byte offset in a
  // VGPR, per-lane 64-bit global address in a VGPR pair. Tracked by ASYNCcnt.
  asm volatile("global_load_async_to_lds_b32 %0, %1, off"
               :
               : "v"(lds_byte_off), "v"(gaddr)
               : "memory");
}

__device__ __forceinline__ void wait_asynccnt0() {
  asm volatile("s_wait_asynccnt 0x0" ::: "memory");
}

__global__ __launch_bounds__(256) void cspn_step_kernel(
    const float* __restrict__ kern,   // (BS, 9, H, W)
    const float* __restrict__ in,     // (BS, 1, H, W)
    const float* __restrict__ in0,    // (BS, 1, H, W)
    float* __restrict__ out,          // (BS, 1, H, W)
    int H, int W) {
  __shared__ float tile[HALO_H * LP];

  const int tx = threadIdx.x;              // 0..31
  const int ty = threadIdx.y;              // 0..7
  const int tid = ty * 32 + tx;            // 0..255
  const int x0 = blockIdx.x * TW;
  const int y0 = blockIdx.y * TH;
  const int b  = blockIdx.z;

  const size_t plane = (size_t)H * W;
  const float* __restrict__ inb = in + (size_t)b * plane;

  // ---- Stage input halo tile (HALO_H x HALO_W) into LDS via async DMA ----
  // LDS row r corresponds to global row y0-1+r; col c to global col x0-1+c.
  for (int c = tid; c < HALO_H * HALO_W; c += 256) {
    const int r  = c / HALO_W;
    const int cc = c - r * HALO_W;
    const int gy = y0 - 1 + r;
    const int gx = x0 - 1 + cc;
    const int li = r * LP + cc;
    if ((unsigned)gy < (unsigned)H && (unsigned)gx < (unsigned)W) {
      const unsigned loff = (unsigned)(uintptr_t)(&tile[li]);
      async_ld_b32_to_lds(loff, inb + (size_t)gy * W + gx);
    } else {
      tile[li] = 0.0f;   // zero padding ('same' conv padding)
    }
  }
  wait_asynccnt0();      // this wave's async copies landed in LDS
  __syncthreads();       // all waves' tiles visible

  // ---- Compute 4 consecutive outputs per thread ----
  const int x = x0 + tx * 4;
  const int y = y0 + ty;
  if (x >= W || y >= H) return;

  const size_t pix = (size_t)b * plane + (size_t)y * W + x;

  // Center tap is replaced by input0 (sparse depth) per the reference.
  const v4f center = __builtin_nontemporal_load((const v4f*)(in0 + pix));

  v4f acc = {0.0f, 0.0f, 0.0f, 0.0f};

#pragma unroll
  for (int i = 0; i < 3; ++i) {
#pragma unroll
    for (int j = 0; j < 3; ++j) {
      const int k = i * 3 + j;
      // Per-pixel weight plane k: single-use stream -> non-temporal b128.
      const v4f wv = __builtin_nontemporal_load(
          (const v4f*)(kern + ((size_t)(b * 9 + k)) * plane +
                       (size_t)y * W + x));
      if (k == 4) {
        acc += wv * center;
      } else {
        const float* trow = &tile[(ty + i) * LP + tx * 4 + j];
        const v4f pv = {trow[0], trow[1], trow[2], trow[3]};
        acc += wv * pv;
      }
    }
  }

  __builtin_nontemporal_store(acc, (v4f*)(out + pix));
}

extern "C" void kernel_launch(void* const* d_in, const int* in_sizes, int n_in,
                              void* d_out, int out_size, void* d_ws,
                              size_t ws_size, hipStream_t stream) {
  (void)n_in; (void)d_ws; (void)ws_size; (void)out_size;

  const float* kern = (const float*)d_in[0];  // (bs, 9, h, w)
  // d_in[1] = iter_weight: unused (reference returns cspn_output directly).
  const float* in   = (const float*)d_in[2];  // (bs, 1, h, w)
  const float* in0  = (const float*)d_in[3];  // (bs, 1, h, w)
  float* out = (float*)d_out;

  const int H = 352;
  const int W = 1216;
  const int BS = in_sizes[2] / (H * W);       // = 8

  dim3 block(32, 8, 1);
  dim3 grid((W + TW - 1) / TW, (H + TH - 1) / TH, BS);
  cspn_step_kernel<<<grid, block, 0, stream>>>(kern, in, in0, out, H, W);
}